// OuterProductMean_26783416058535
// MI455X (gfx1250) — compile-verified
//
#include <hip/hip_runtime.h>

// OuterProductMean for MI455X (gfx1250, wave32).
// Factored algorithm (avoids the 17-GFLOP c*c GEMM formulation):
//   A: a2/b2[l,c] = mean_n( LN(m[n,l,:]) @ W{i,j} ) + b{i,j}      (reads 64 MB)
//   B: U[i,d,z]   = sum_c a2[i,c] * Wo[c,d,z]                      (4 MB, L2)
//   C: out[i,j,z] = sum_d b2[j,d] * U[i,d,z] + bo[z]               (writes 128 MB, WMMA)

typedef __attribute__((ext_vector_type(2))) float v2f;
typedef __attribute__((ext_vector_type(8))) float v8f;

#define L_DIM 512
#define N_DIM 128
#define CM    256
#define CHID  16
#define CZ    128
#define LN_EPS 1e-5f

// ---------------------------------------------------------------------------
// Stage A: LayerNorm over c_m, project to c=16 with Wi and Wj, mean over N.
// One block per l (256 threads = 8 waves; thread t owns column t of c_m).
// ---------------------------------------------------------------------------
__global__ __launch_bounds__(256)
void opm_stageA(const float* __restrict__ m, const float* __restrict__ gamma,
                const float* __restrict__ beta,
                const float* __restrict__ Wi, const float* __restrict__ bi,
                const float* __restrict__ Wj, const float* __restrict__ bj,
                float* __restrict__ a2, float* __restrict__ b2)
{
    __shared__ float sred[16];            // 8 waves x (sum, sumsq)
    __shared__ float smat[256 * CHID];    // 16 KB final reduction buffer

    const int l    = blockIdx.x;
    const int t    = threadIdx.x;
    const int lane = t & 31;
    const int wave = t >> 5;

    const float g  = gamma[t];
    const float be = beta[t];

    float wi[CHID], wj[CHID];
#pragma unroll
    for (int c = 0; c < CHID; ++c) {
        wi[c] = Wi[t * CHID + c];
        wj[c] = Wj[t * CHID + c];
    }

    float accA[CHID], accB[CHID];
#pragma unroll
    for (int c = 0; c < CHID; ++c) { accA[c] = 0.f; accB[c] = 0.f; }

    for (int n = 0; n < N_DIM; ++n) {
        const float x = m[((size_t)n * L_DIM + l) * CM + t];
        if (n + 1 < N_DIM)
            __builtin_prefetch(&m[((size_t)(n + 1) * L_DIM + l) * CM + t], 0, 0);

        // block-wide reduction of (sum x, sum x^2): wave shuffle then LDS
        float s = x, q = x * x;
#pragma unroll
        for (int off = 16; off > 0; off >>= 1) {
            s += __shfl_xor(s, off, 32);
            q += __shfl_xor(q, off, 32);
        }
        if (lane == 0) { sred[wave * 2] = s; sred[wave * 2 + 1] = q; }
        __syncthreads();
        s = 0.f; q = 0.f;
#pragma unroll
        for (int w = 0; w < 8; ++w) { s += sred[w * 2]; q += sred[w * 2 + 1]; }
        __syncthreads();   // sred reusable next iteration

        const float mu  = s * (1.f / CM);
        const float var = q * (1.f / CM) - mu * mu;
        const float y   = (x - mu) * rsqrtf(var + LN_EPS) * g + be;

#pragma unroll
        for (int c = 0; c < CHID; ++c) {
            accA[c] += y * wi[c];
            accB[c] += y * wj[c];
        }
    }

    // reduce accA across the 256 threads (k dimension of the projection)
#pragma unroll
    for (int c = 0; c < CHID; ++c) smat[t * CHID + c] = accA[c];
    __syncthreads();
    for (int sft = 128; sft >= 1; sft >>= 1) {
        if (t < sft) {
#pragma unroll
            for (int c = 0; c < CHID; ++c)
                smat[t * CHID + c] += smat[(t + sft) * CHID + c];
        }
        __syncthreads();
    }
    if (t < CHID) a2[l * CHID + t] = smat[t] * (1.f / N_DIM) + bi[t];
    __syncthreads();

    // reduce accB
#pragma unroll
    for (int c = 0; c < CHID; ++c) smat[t * CHID + c] = accB[c];
    __syncthreads();
    for (int sft = 128; sft >= 1; sft >>= 1) {
        if (t < sft) {
#pragma unroll
            for (int c = 0; c < CHID; ++c)
                smat[t * CHID + c] += smat[(t + sft) * CHID + c];
        }
        __syncthreads();
    }
    if (t < CHID) b2[l * CHID + t] = smat[t] * (1.f / N_DIM) + bj[t];
}

// ---------------------------------------------------------------------------
// Stage B: U[i,d,z] = sum_c a2[i,c] * Wo[(c*16+d)*128 + z].  One block per i.
// ---------------------------------------------------------------------------
__global__ __launch_bounds__(256)
void opm_stageB(const float* __restrict__ a2, const float* __restrict__ Wo,
                float* __restrict__ U)
{
    __shared__ float a[CHID];
    const int i = blockIdx.x;
    const int t = threadIdx.x;
    if (t < CHID) a[t] = a2[i * CHID + t];
    __syncthreads();

    for (int e = t; e < CHID * CZ; e += 256) {
        const int d = e >> 7;          // CZ = 128
        const int z = e & (CZ - 1);
        float s = 0.f;
#pragma unroll
        for (int c = 0; c < CHID; ++c)
            s += a[c] * Wo[((size_t)(c * CHID + d)) * CZ + z];
        U[(size_t)i * (CHID * CZ) + e] = s;
    }
}

// ---------------------------------------------------------------------------
// Stage C: out[i, j, z] = sum_d b2[j,d] * U[i,d,z] + bo[z]   (WMMA f32 16x16x4)
// Grid: (i, j-tile).  4 waves per block; wave w covers z-tiles w and w+4.
// A (16x4 f32): lanes 0-15 hold (M=lane, K=k0..k0+1); lanes 16-31 K=k0+2..k0+3.
// B (4x16 f32): VGPR r: lanes 0-15 row K=k0+r, lanes 16-31 row K=k0+r+2.
// C/D (16x16 f32): VGPR r: lanes 0-15 row M=r, lanes 16-31 row M=r+8.
// ---------------------------------------------------------------------------
__global__ __launch_bounds__(128)
void opm_stageC(const float* __restrict__ b2, const float* __restrict__ U,
                const float* __restrict__ bo, float* __restrict__ out)
{
    const int i     = blockIdx.x;
    const int jt    = blockIdx.y;
    const int lane  = threadIdx.x & 31;
    const int wave  = threadIdx.x >> 5;
    const int jbase = jt * 16;
    const int mrow  = lane & 15;
    const int ncol  = lane & 15;
    const int khalf = (lane >> 4) * 2;     // upper half-wave owns K+2, K+3

    const float* __restrict__ Ui = U + (size_t)i * (CHID * CZ);

    // A fragments: b2 j-tile, one v2f per K-step of 4 (K total = 16)
    v2f afrag[4];
#pragma unroll
    for (int k = 0; k < 4; ++k) {
        const float* p = &b2[(jbase + mrow) * CHID + k * 4 + khalf];
        afrag[k].x = p[0];
        afrag[k].y = p[1];
    }

#pragma unroll
    for (int zi = 0; zi < 2; ++zi) {
        const int zbase = (wave + zi * 4) * 16;

        v8f acc = {};
#pragma unroll
        for (int k = 0; k < 4; ++k) {
            v2f bfrag;
            bfrag.x = Ui[(size_t)(k * 4 + 0 + khalf) * CZ + zbase + ncol];
            bfrag.y = Ui[(size_t)(k * 4 + 1 + khalf) * CZ + zbase + ncol];
            // 8 args: (neg_a, A, neg_b, B, c_mod, C, reuse_a, reuse_b)
            acc = __builtin_amdgcn_wmma_f32_16x16x4_f32(
                false, afrag[k], false, bfrag, (short)0, acc, false, false);
        }

        const float bz = bo[zbase + ncol];
        const int   m0 = (lane >> 4) * 8;
#pragma unroll
        for (int r = 0; r < 8; ++r) {
            const int j = jbase + m0 + r;
            out[((size_t)i * L_DIM + j) * CZ + zbase + ncol] = acc[r] + bz;
        }
    }
}

// ---------------------------------------------------------------------------
extern "C" void kernel_launch(void* const* d_in, const int* in_sizes, int n_in,
                              void* d_out, int out_size, void* d_ws, size_t ws_size,
                              hipStream_t stream)
{
    const float* m     = (const float*)d_in[0];
    const float* gamma = (const float*)d_in[1];
    const float* beta  = (const float*)d_in[2];
    const float* Wi    = (const float*)d_in[3];
    const float* bi    = (const float*)d_in[4];
    const float* Wj    = (const float*)d_in[5];
    const float* bj    = (const float*)d_in[6];
    const float* Wo    = (const float*)d_in[7];
    const float* bo    = (const float*)d_in[8];
    float* out = (float*)d_out;

    float* a2 = (float*)d_ws;                 // 512*16 f32
    float* b2 = a2 + L_DIM * CHID;            // 512*16 f32
    float* U  = b2 + L_DIM * CHID;            // 512*16*128 f32 (4 MB)

    opm_stageA<<<dim3(L_DIM), dim3(256), 0, stream>>>(m, gamma, beta, Wi, bi, Wj, bj, a2, b2);
    opm_stageB<<<dim3(L_DIM), dim3(256), 0, stream>>>(a2, Wo, U);
    opm_stageC<<<dim3(L_DIM, 32), dim3(128), 0, stream>>>(b2, U, bo, out);
}